// SpaceToDepth_11811160064501
// MI455X (gfx1250) — compile-verified
//
#include <hip/hip_runtime.h>
#include <hip/hip_bf16.h>

#ifndef __has_builtin
#define __has_builtin(x) 0
#endif

#if defined(__gfx1250__) && __has_builtin(__builtin_amdgcn_global_load_async_to_lds_b128) && __has_builtin(__builtin_amdgcn_s_wait_asynccnt)
#define S2D_USE_ASYNC_LDS 1
#else
#define S2D_USE_ASYNC_LDS 0
#endif

typedef __attribute__((ext_vector_type(4))) float f32x4;
typedef __attribute__((ext_vector_type(4))) int   i32x4;
typedef __attribute__((address_space(1))) i32x4*  gvec_ptr;  // global (AS1)
typedef __attribute__((address_space(3))) i32x4*  lvec_ptr;  // LDS (AS3)

// Shapes fixed by the reference:
//   x0: [16, 64, 256, 256] f32, x1: [16] f32
//   out: [16, 256, 128, 128] f32 (++ x1 appended flat)
#define S2D_B   16
#define S2D_C   64
#define S2D_H   256
#define S2D_W   256
#define S2D_HO  128
#define S2D_WO  128
#define S2D_OUT_ELEMS ((size_t)S2D_B * 4 * S2D_C * S2D_HO * S2D_WO)

// One thread: 8 consecutive input floats (two b128 async loads via LDS)
// -> 4 floats to the (w_in=0) plane and 4 floats to the (w_in=1) plane
// (two b128 stores). Fully coalesced in both directions.
__global__ __launch_bounds__(256) void SpaceToDepth_kernel(
    const float* __restrict__ x0,
    const float* __restrict__ x1,
    float* __restrict__ out) {
  const unsigned t = blockIdx.x * 256u + threadIdx.x;

  const unsigned j = t & 31u;           // 8-float chunk within the 256-wide row
  const unsigned h = (t >> 5) & 255u;   // input row
  const unsigned c = (t >> 13) & 63u;   // input channel
  const unsigned b = t >> 19;           // batch

  const float* src =
      x0 + ((size_t)(b * S2D_C + c) * S2D_H + h) * S2D_W + (size_t)j * 8u;

  f32x4 a0, a1;
#if S2D_USE_ASYNC_LDS
  // Stage the 32B through LDS with the CDNA5 async data mover
  // (ASYNCcnt-tracked global_load_async_to_lds_b128), then read back.
  __shared__ float lds[256 * 8];
  float* slot = &lds[(unsigned)threadIdx.x * 8u];
  float* srcnc = const_cast<float*>(src);
  __builtin_amdgcn_global_load_async_to_lds_b128(
      (gvec_ptr)(srcnc), (lvec_ptr)(slot), 0, 0);
  __builtin_amdgcn_global_load_async_to_lds_b128(
      (gvec_ptr)(srcnc + 4), (lvec_ptr)(slot + 4), 0, 0);
  __builtin_amdgcn_s_wait_asynccnt(0);
  __asm__ volatile("" ::: "memory");
  a0 = *(const f32x4*)(slot);
  a1 = *(const f32x4*)(slot + 4);
#else
  a0 = *(const f32x4*)(src);
  a1 = *(const f32x4*)(src + 4);
#endif

  const unsigned ho  = h >> 1;
  const unsigned hin = h & 1u;
  // depth index d = h_in*128 + w_in*64 + c
  const unsigned d0 = hin * 128u + c;        // w_in = 0 plane
  const unsigned d1 = d0 + 64u;              // w_in = 1 plane
  const unsigned wo = j * 4u;

  float* p0 = out + ((size_t)(b * 256u + d0) * S2D_HO + ho) * S2D_WO + wo;
  float* p1 = out + ((size_t)(b * 256u + d1) * S2D_HO + ho) * S2D_WO + wo;

  f32x4 even = {a0.x, a0.z, a1.x, a1.z};     // w = 8j, 8j+2, 8j+4, 8j+6
  f32x4 odd  = {a0.y, a0.w, a1.y, a1.w};     // w = 8j+1, 8j+3, 8j+5, 8j+7
  *(f32x4*)p0 = even;
  *(f32x4*)p1 = odd;

  // Passthrough of x1 (16 floats appended after the image output).
  if (blockIdx.x == 0 && threadIdx.x < 16u) {
    out[S2D_OUT_ELEMS + threadIdx.x] = x1[threadIdx.x];
  }
}

extern "C" void kernel_launch(void* const* d_in, const int* in_sizes, int n_in,
                              void* d_out, int out_size, void* d_ws, size_t ws_size,
                              hipStream_t stream) {
  (void)in_sizes; (void)n_in; (void)out_size; (void)d_ws; (void)ws_size;
  const float* x0 = (const float*)d_in[0];
  const float* x1 = (const float*)d_in[1];
  float* out = (float*)d_out;

  // total threads = 16*64*256*32 = 8,388,608 -> 32768 blocks of 256
  const unsigned total = S2D_B * S2D_C * S2D_H * (S2D_W / 8);
  dim3 grid(total / 256u), block(256u);
  SpaceToDepth_kernel<<<grid, block, 0, stream>>>(x0, x1, out);
}